// BidPrefix_28432683499802
// MI455X (gfx1250) — compile-verified
//
#include <hip/hip_runtime.h>

#define EPS_F 1e-7f
#define ROW_LEN 512          // S, fixed by the reference
#define ELEMS_PER_LANE 16    // 512 / wave32
#define WAVES_PER_BLOCK 8    // 256 threads

typedef float v4f __attribute__((ext_vector_type(4)));

// One wave32 per row. Each lane owns 16 contiguous floats (4 x b128 NT loads).
// Three masked product accumulators give cp[bid], cp[mp], cp[mp+1] exactly as
// the reference computes them; butterfly multiply-reduce across the wave.
__global__ __launch_bounds__(256) void BidPrefix_kernel(
    const float* __restrict__ x,
    const int*   __restrict__ bid_info,
    float* __restrict__ survival,
    float* __restrict__ rate_last,
    int B)
{
    const int wave = threadIdx.x >> 5;
    const int lane = threadIdx.x & 31;
    const int row  = blockIdx.x * WAVES_PER_BLOCK + wave;
    if (row >= B) return;  // wave-uniform

    // bid_info is (B,2): [:,0] = mp, [:,1] = bid. Wave-uniform scalars.
    const int mp  = bid_info[(size_t)row * 2 + 0];
    const int bid = bid_info[(size_t)row * 2 + 1];

    const int t_bid = bid;      // survival needs j <  bid
    const int t_mp  = mp;       // cp[mp]   needs j <  mp
    const int t_mp1 = mp + 1;   // cp[mp+1] needs j <= mp
    const int t_max = (t_bid > t_mp1) ? t_bid : t_mp1;

    const int base = lane * ELEMS_PER_LANE;

    float accBid = 1.0f, accMp = 1.0f, accMp1 = 1.0f;

    // Lanes whose whole chunk is past every threshold issue no VMEM at all:
    // expected saving ~1/3 of the 128 MB stream.
    if (base < t_max) {
        const v4f* xr = (const v4f*)(x + (size_t)row * ROW_LEN + base);
        float v[ELEMS_PER_LANE];
        #pragma unroll
        for (int q = 0; q < 4; ++q) {
            // Single-pass 128 MB stream, zero reuse -> NT temporal hint so we
            // don't thrash L2/WGP$ (gfx1250 TH cache-control path).
            v4f f = __builtin_nontemporal_load(xr + q);   // global_load_b128 th:TH_LOAD_NT
            v[q * 4 + 0] = f.x;
            v[q * 4 + 1] = f.y;
            v[q * 4 + 2] = f.z;
            v[q * 4 + 3] = f.w;
        }
        #pragma unroll
        for (int i = 0; i < ELEMS_PER_LANE; ++i) {
            const int   g   = base + i;
            const float val = v[i];
            accBid *= (g < t_bid) ? val : 1.0f;   // v_cmp + v_cndmask + v_mul (VOPD-pairable)
            accMp  *= (g < t_mp ) ? val : 1.0f;
            accMp1 *= (g < t_mp1) ? val : 1.0f;
        }
    }

    // Wave32 butterfly multiply-reduction (all lanes reconverged here).
    #pragma unroll
    for (int m = 16; m >= 1; m >>= 1) {
        accBid *= __shfl_xor(accBid, m, 32);
        accMp  *= __shfl_xor(accMp,  m, 32);
        accMp1 *= __shfl_xor(accMp1, m, 32);
    }

    if (lane == 0) {
        const float rate = (mp == 0) ? EPS_F : (accMp - accMp1);
        __builtin_nontemporal_store(accBid, survival + row);   // write-once outputs
        __builtin_nontemporal_store(rate,  rate_last + row);
    }
}

extern "C" void kernel_launch(void* const* d_in, const int* in_sizes, int n_in,
                              void* d_out, int out_size, void* d_ws, size_t ws_size,
                              hipStream_t stream) {
    const float* x        = (const float*)d_in[0];   // (B, 512) f32
    const int*   bid_info = (const int*)d_in[1];     // (B, 2)   i32

    const int B = in_sizes[1] / 2;

    float* out       = (float*)d_out;
    float* survival  = out;        // first B floats
    float* rate_last = out + B;    // next  B floats

    const int blocks = (B + WAVES_PER_BLOCK - 1) / WAVES_PER_BLOCK;
    BidPrefix_kernel<<<blocks, 256, 0, stream>>>(x, bid_info, survival, rate_last, B);
}